// Model_19112604467649
// MI455X (gfx1250) — compile-verified
//
#include <hip/hip_runtime.h>
#include <math.h>

// ---------------------------------------------------------------------------
// MI455X (gfx1250) implementation of the dual-Mamba segmentation block.
// All GEMMs use V_WMMA_F32_16X16X4_F32 (full fp32, wave32, 16x16 tile/wave).
// Layout convention: activations stored [p, ch] row-major, p = b*4096 + h*64 + w.
// ---------------------------------------------------------------------------

typedef __attribute__((ext_vector_type(2))) float v2f;
typedef __attribute__((ext_vector_type(8))) float v8f;

#define HWSZ 4096
#define CCH  128

__device__ __forceinline__ float silu_f(float x)     { return x / (1.f + __expf(-x)); }
__device__ __forceinline__ float softplus_f(float x) { return (x > 20.f) ? x : log1pf(__expf(x)); }

// ---------------------------------------------------------------------------
// Generic NT GEMM:  D[m,n] = sum_k A[m,k] * W[n,k] (+ bias[n])
// Block = 4 waves; wave w computes 16x16 tile (blockIdx.x, blockIdx.y*4 + w).
// Requires: K % 4 == 0 and M % 16 == 0 (true at every call site).
// Ragged N handled by CLAMPING the W row for loads (garbage only reaches
// output columns n >= N, which the store guard discards) -> the inner loop is
// branch-free: 2x global_load_b64 + v_wmma per k-step, no EXEC manipulation.
// Frag layouts per ISA 7.12.2:
//   A 16x4 : lanes 0-15 hold K=k0,k0+1 ; lanes 16-31 hold K=k0+2,k0+3
//   B 4x16 : same K split across lane halves, N = lane&15
//   C/D    : VGPR v -> row M=v (lanes 0-15) / M=v+8 (lanes 16-31), N = lane&15
// ---------------------------------------------------------------------------
__global__ __launch_bounds__(128) void gemm_nt_wmma(
    const float* __restrict__ A, const float* __restrict__ W,
    const float* __restrict__ bias, float* __restrict__ D,
    int M, int N, int K, int lda, int ldw, int ldd)
{
    const int lane = threadIdx.x;                 // 0..31
    const int half = lane >> 4;
    const int li   = lane & 15;
    const int m  = blockIdx.x * 16 + li;                       // always < M
    const int n  = (blockIdx.y * 4 + threadIdx.y) * 16 + li;   // may exceed N
    const int nc = (n < N) ? n : (N - 1);                      // clamped for loads
    const float* arow = A + (size_t)m  * lda + 2 * half;
    const float* wrow = W + (size_t)nc * ldw + 2 * half;

    v8f acc = {0.f, 0.f, 0.f, 0.f, 0.f, 0.f, 0.f, 0.f};
    for (int k0 = 0; k0 < K; k0 += 4) {
        v2f a = *(const v2f*)(arow + k0);
        v2f b = *(const v2f*)(wrow + k0);
        acc = __builtin_amdgcn_wmma_f32_16x16x4_f32(
                  /*neg_a=*/false, a, /*neg_b=*/false, b,
                  /*c_mod=*/(short)0, acc, /*reuse_a=*/false, /*reuse_b=*/false);
    }
    const float bv = (bias != nullptr) ? bias[nc] : 0.f;
    if (n < N) {
#pragma unroll
        for (int v = 0; v < 8; ++v) {
            const int mm = blockIdx.x * 16 + v + half * 8;
            D[(size_t)mm * ldd + n] = acc[v] + bv;
        }
    }
}

// x (B,C,H,W) -> xT [p, c]
__global__ void transpose_in(const float* __restrict__ x, float* __restrict__ xT)
{
    int i = blockIdx.x * 256 + threadIdx.x;          // i = p*128 + c
    if (i >= 2 * HWSZ * CCH) return;
    int c = i & 127, p = i >> 7, b = p >> 12, hw = p & 4095;
    xT[i] = x[((size_t)(b * CCH + c)) * HWSZ + hw];
}

// GroupNorm stats: grid = B*4 (4 groups of 32 channels), X is [p,128]
__global__ __launch_bounds__(256) void gn_stats(const float* __restrict__ X,
                                                float* __restrict__ stats)
{
    const int b = blockIdx.x >> 2, g = blockIdx.x & 3;
    float s = 0.f, ss = 0.f;
    for (int i = threadIdx.x; i < HWSZ * 32; i += 256) {
        int hw = i >> 5, c = i & 31;
        float v = X[((size_t)(b * HWSZ + hw)) * CCH + g * 32 + c];
        s += v; ss += v * v;
    }
    __shared__ float rs[256], rq[256];
    rs[threadIdx.x] = s; rq[threadIdx.x] = ss;
    __syncthreads();
    for (int o = 128; o > 0; o >>= 1) {
        if ((int)threadIdx.x < o) {
            rs[threadIdx.x] += rs[threadIdx.x + o];
            rq[threadIdx.x] += rq[threadIdx.x + o];
        }
        __syncthreads();
    }
    if (threadIdx.x == 0) {
        const float inv = 1.f / (float)(HWSZ * 32);
        float m = rs[0] * inv;
        float var = rq[0] * inv - m * m;
        stats[blockIdx.x * 2]     = m;
        stats[blockIdx.x * 2 + 1] = rsqrtf(var + 1e-5f);
    }
}

// out = silu(GN(X)) (+ addend)   — elementwise over [p,128]
__global__ void gn_apply(const float* __restrict__ X, const float* __restrict__ stats,
                         const float* __restrict__ gamma, const float* __restrict__ beta,
                         const float* __restrict__ addend, float* __restrict__ out)
{
    int i = blockIdx.x * 256 + threadIdx.x;
    if (i >= 2 * HWSZ * CCH) return;
    int c = i & 127, p = i >> 7, b = p >> 12, grp = c >> 5;
    int sidx = (b * 4 + grp) * 2;
    float v = (X[i] - stats[sidx]) * stats[sidx + 1] * gamma[c] + beta[c];
    v = silu_f(v);
    if (addend) v += addend[i];
    out[i] = v;
}

// Spatial causal depthwise conv (K=4) + silu: xz [p,512] (x-half cols 0..255) -> xc [p,256]
__global__ void conv1d_spatial(const float* __restrict__ xz, const float* __restrict__ cw,
                               const float* __restrict__ cb, float* __restrict__ xc)
{
    int t = blockIdx.x * 256 + threadIdx.x;          // p*256 + d
    if (t >= 8192 * 256) return;
    int d = t & 255, p = t >> 8, b = p >> 12, l = p & 4095;
    float acc = cb[d];
#pragma unroll
    for (int k = 0; k < 4; ++k) {
        int ls = l - 3 + k;
        if (ls >= 0) acc += xz[((size_t)((b << 12) + ls)) * 512 + d] * cw[d * 4 + k];
    }
    xc[t] = silu_f(acc);
}

// Spectral causal depthwise conv (L=4) + silu: xz2 [row,128] -> xc2 [row,64]
__global__ void conv1d_spectral(const float* __restrict__ xz2, const float* __restrict__ cw,
                                const float* __restrict__ cb, float* __restrict__ xc2)
{
    int t = blockIdx.x * 256 + threadIdx.x;          // row*64 + d
    if (t >= 32768 * 64) return;
    int d = t & 63, row = t >> 6, j = row & 3, q = row >> 2;
    float acc = cb[d];
#pragma unroll
    for (int k = 0; k < 4; ++k) {
        int js = j - 3 + k;
        if (js >= 0) acc += xz2[((size_t)(q * 4 + js)) * 128 + d] * cw[d * 4 + k];
    }
    xc2[t] = silu_f(acc);
}

// Spatial selective scan, L=4096. One thread per (b,d); h[16] lives in VGPRs.
// proj layout per row (ld 48): dt[0..8) | B[8..24) | C[24..40). Gate silu(z) fused.
__global__ __launch_bounds__(256) void scan_spatial(
    const float* __restrict__ deltaraw,  // [8192,256]
    const float* __restrict__ xc,        // [8192,256]
    const float* __restrict__ proj,      // [8192,48]
    const float* __restrict__ xz,        // [8192,512] (z = cols 256..511)
    const float* __restrict__ Alog,      // [256,16]
    const float* __restrict__ Dvec,      // [256]
    float* __restrict__ yg)              // [8192,256]
{
    const int b = blockIdx.x;            // 0..1
    const int d = threadIdx.x;           // 0..255
    __shared__ float sBC[32];
    float A[16], h[16];
#pragma unroll
    for (int n = 0; n < 16; ++n) { A[n] = -__expf(Alog[d * 16 + n]); h[n] = 0.f; }
    const float Dd = Dvec[d];

    for (int l = 0; l < 4096; ++l) {
        const int p = (b << 12) + l;
        __syncthreads();
        if (threadIdx.x < 32) sBC[threadIdx.x] = proj[(size_t)p * 48 + 8 + threadIdx.x];
        __syncthreads();
        const float delta = softplus_f(deltaraw[(size_t)p * 256 + d]);
        const float xcv   = xc[(size_t)p * 256 + d];
        float acc = 0.f;
#pragma unroll
        for (int n = 0; n < 16; ++n) {
            const float dA = __expf(delta * A[n]);
            h[n] = dA * h[n] + (delta * sBC[n]) * xcv;
            acc += h[n] * sBC[16 + n];
        }
        const float zv = xz[(size_t)p * 512 + 256 + d];
        yg[(size_t)p * 256 + d] = (acc + Dd * xcv) * silu_f(zv);
    }
}

// Spectral selective scan, L=4. One thread per (q,d); dt GEMM (K=2) folded in.
// proj2 per row (ld 48): dt[0..2) | B[2..18) | C[18..34).
__global__ void scan_spectral(
    const float* __restrict__ xz2,   // [32768,128] (z = cols 64..127)
    const float* __restrict__ xc2,   // [32768,64]
    const float* __restrict__ proj2, // [32768,48]
    const float* __restrict__ dtw,   // [64,2]
    const float* __restrict__ dtb,   // [64]
    const float* __restrict__ Alog,  // [64,16]
    const float* __restrict__ Dvec,  // [64]
    float* __restrict__ yg2)         // [32768,64]
{
    int t = blockIdx.x * 256 + threadIdx.x;          // q*64 + d
    if (t >= 8192 * 64) return;
    const int q = t >> 6, d = t & 63;
    float A[16], h[16];
#pragma unroll
    for (int n = 0; n < 16; ++n) { A[n] = -__expf(Alog[d * 16 + n]); h[n] = 0.f; }
    const float w0 = dtw[d * 2], w1 = dtw[d * 2 + 1], bb = dtb[d], Dd = Dvec[d];

#pragma unroll
    for (int j = 0; j < 4; ++j) {
        const int row = q * 4 + j;
        const float* pr = proj2 + (size_t)row * 48;
        const float delta = softplus_f(pr[0] * w0 + pr[1] * w1 + bb);
        const float xcv = xc2[(size_t)row * 64 + d];
        float acc = 0.f;
#pragma unroll
        for (int n = 0; n < 16; ++n) {
            h[n] = __expf(delta * A[n]) * h[n] + (delta * pr[2 + n]) * xcv;
            acc += h[n] * pr[18 + n];
        }
        const float zv = xz2[(size_t)row * 128 + 64 + d];
        yg2[(size_t)row * 64 + d] = (acc + Dd * xcv) * silu_f(zv);
    }
}

// feat = spa*w0 + spe*w1 + h1, w = softmax(fuse_w)
__global__ void fuse_kernel(const float* __restrict__ spa, const float* __restrict__ spe,
                            const float* __restrict__ h1, const float* __restrict__ fw,
                            float* __restrict__ feat)
{
    int i = blockIdx.x * 256 + threadIdx.x;
    if (i >= 2 * HWSZ * CCH) return;
    float e0 = __expf(fw[0]), e1 = __expf(fw[1]);
    float inv = 1.f / (e0 + e1);
    feat[i] = spa[i] * (e0 * inv) + spe[i] * (e1 * inv) + h1[i];
}

// depthwise 3x3 conv on feat ([p,128] viewed as image), zero padding
__global__ void dwconv3x3(const float* __restrict__ feat, const float* __restrict__ dww,
                          float* __restrict__ edge)
{
    int i = blockIdx.x * 256 + threadIdx.x;          // p*128 + c
    if (i >= 2 * HWSZ * CCH) return;
    int c = i & 127, p = i >> 7, b = p >> 12, hw = p & 4095;
    int h = hw >> 6, w = hw & 63;
    float acc = 0.f;
#pragma unroll
    for (int dy = -1; dy <= 1; ++dy)
#pragma unroll
        for (int dx = -1; dx <= 1; ++dx) {
            int hh = h + dy, ww = w + dx;
            if (hh >= 0 && hh < 64 && ww >= 0 && ww < 64)
                acc += feat[((size_t)((b << 12) + (hh << 6) + ww)) * CCH + c]
                       * dww[c * 9 + (dy + 1) * 3 + (dx + 1)];
        }
    edge[i] = acc;
}

// edge1[p] = sigmoid(sum_c edge[p,c]*pw_w[c] + pw_b)
__global__ void pw_sigmoid(const float* __restrict__ edge, const float* __restrict__ pww,
                           const float* __restrict__ pwb, float* __restrict__ edge1)
{
    int p = blockIdx.x * 256 + threadIdx.x;
    if (p >= 2 * HWSZ) return;
    float acc = pwb[0];
    const float* row = edge + (size_t)p * CCH;
#pragma unroll 4
    for (int c = 0; c < CCH; ++c) acc += row[c] * pww[c];
    edge1[p] = 1.f / (1.f + __expf(-acc));
}

// out = logits + alpha*(1-edge)*(boxsum3x3(logits)/9 - logits), NCHW output
__global__ void final_combine(const float* __restrict__ logits, const float* __restrict__ edge1,
                              const float* __restrict__ alpha, float* __restrict__ out)
{
    int t = blockIdx.x * 256 + threadIdx.x;          // ((b*10+o)*64+h)*64+w
    if (t >= 2 * 10 * HWSZ) return;
    int w = t & 63, h = (t >> 6) & 63, rest = t >> 12;
    int o = rest % 10, b = rest / 10;
    int pbase = b * HWSZ;
    float l0 = logits[(size_t)(pbase + (h << 6) + w) * 10 + o];
    float loc = 0.f;
#pragma unroll
    for (int dy = -1; dy <= 1; ++dy)
#pragma unroll
        for (int dx = -1; dx <= 1; ++dx) {
            int hh = h + dy, ww = w + dx;
            if (hh >= 0 && hh < 64 && ww >= 0 && ww < 64)
                loc += logits[(size_t)(pbase + (hh << 6) + ww) * 10 + o];
        }
    loc *= (1.f / 9.f);
    float e = edge1[pbase + (h << 6) + w];
    out[t] = l0 + alpha[0] * (1.f - e) * (loc - l0);
}

// ---------------------------------------------------------------------------
extern "C" void kernel_launch(void* const* d_in, const int* in_sizes, int n_in,
                              void* d_out, int out_size, void* d_ws, size_t ws_size,
                              hipStream_t stream)
{
    const float* x          = (const float*)d_in[0];
    const float* pe_w       = (const float*)d_in[1];
    const float* pe_b       = (const float*)d_in[2];
    const float* pe_gn_g    = (const float*)d_in[3];
    const float* pe_gn_b    = (const float*)d_in[4];
    const float* spa_in_w   = (const float*)d_in[5];
    const float* spa_conv_w = (const float*)d_in[6];
    const float* spa_conv_b = (const float*)d_in[7];
    const float* spa_xproj_w= (const float*)d_in[8];
    const float* spa_dt_w   = (const float*)d_in[9];
    const float* spa_dt_b   = (const float*)d_in[10];
    const float* spa_Alog   = (const float*)d_in[11];
    const float* spa_D      = (const float*)d_in[12];
    const float* spa_out_w  = (const float*)d_in[13];
    const float* spa_gn_g   = (const float*)d_in[14];
    const float* spa_gn_b   = (const float*)d_in[15];
    const float* spe_in_w   = (const float*)d_in[16];
    const float* spe_conv_w = (const float*)d_in[17];
    const float* spe_conv_b = (const float*)d_in[18];
    const float* spe_xproj_w= (const float*)d_in[19];
    const float* spe_dt_w   = (const float*)d_in[20];
    const float* spe_dt_b   = (const float*)d_in[21];
    const float* spe_Alog   = (const float*)d_in[22];
    const float* spe_D      = (const float*)d_in[23];
    const float* spe_out_w  = (const float*)d_in[24];
    const float* spe_gn_g   = (const float*)d_in[25];
    const float* spe_gn_b   = (const float*)d_in[26];
    const float* fuse_w     = (const float*)d_in[27];
    const float* cls_w      = (const float*)d_in[28];
    const float* cls_b      = (const float*)d_in[29];
    const float* ref_dw_w   = (const float*)d_in[30];
    const float* ref_gn_g   = (const float*)d_in[31];
    const float* ref_gn_b   = (const float*)d_in[32];
    const float* ref_pw_w   = (const float*)d_in[33];
    const float* ref_pw_b   = (const float*)d_in[34];
    const float* alpha      = (const float*)d_in[35];

    float* out = (float*)d_out;
    float* ws  = (float*)d_ws;

    const size_t U = 524288;                 // half of 1M-float unit
    float* h1       = ws + 0 * U;            // [8192,128]
    float* xz       = ws + 2 * U;            // [8192,512]; also xT temp & xz2 [32768,128]
    float* xc       = ws + 10 * U;           // [8192,256]; also xc2 [32768,64]
    float* proj     = ws + 14 * U;           // [8192,48]; also proj2 [32768,48] (spans into deltaraw)
    float* deltaraw = ws + 15 * U;           // [8192,256]
    float* yg       = ws + 19 * U;           // [8192,256]; also yg2 [32768,64]
    float* y2img    = ws + 23 * U;           // [8192,128]
    float* spa      = ws + 25 * U;           // [8192,128]
    float* speimg   = ws + 27 * U;           // [32768,32] == [8192,128]
    float* spe      = ws + 29 * U;           // [8192,128]
    float* feat     = ws + 31 * U;           // [8192,128]
    float* edge     = ws + 33 * U;           // [8192,128]
    float* logits   = ws + 35 * U;           // [8192,10]
    float* edge1    = logits + 81920;        // [8192]
    float* stats    = edge1 + 8192;          // [16]

    const dim3 BW(32, 4);                    // 4 waves per block, 1 N-tile each
    const dim3 B256(256);

    // ---- patch-embed: h1 = silu(GN(x @ pe_w^T + pe_b)) ----
    transpose_in<<<4096, B256, 0, stream>>>(x, xz);                    // xT in xz region
    gemm_nt_wmma<<<dim3(512, 2), BW, 0, stream>>>(xz, pe_w, pe_b, h1,
                                                  8192, 128, 128, 128, 128, 128);
    gn_stats<<<8, B256, 0, stream>>>(h1, stats);
    gn_apply<<<4096, B256, 0, stream>>>(h1, stats, pe_gn_g, pe_gn_b, nullptr, h1);

    // ---- spatial mamba ----
    gemm_nt_wmma<<<dim3(512, 8), BW, 0, stream>>>(h1, spa_in_w, nullptr, xz,
                                                  8192, 512, 128, 128, 128, 512);
    conv1d_spatial<<<8192, B256, 0, stream>>>(xz, spa_conv_w, spa_conv_b, xc);
    gemm_nt_wmma<<<dim3(512, 1), BW, 0, stream>>>(xc, spa_xproj_w, nullptr, proj,
                                                  8192, 40, 256, 256, 256, 48);
    gemm_nt_wmma<<<dim3(512, 4), BW, 0, stream>>>(proj, spa_dt_w, spa_dt_b, deltaraw,
                                                  8192, 256, 8, 48, 8, 256);
    scan_spatial<<<2, B256, 0, stream>>>(deltaraw, xc, proj, xz, spa_Alog, spa_D, yg);
    gemm_nt_wmma<<<dim3(512, 2), BW, 0, stream>>>(yg, spa_out_w, nullptr, y2img,
                                                  8192, 128, 256, 256, 256, 128);
    gn_stats<<<8, B256, 0, stream>>>(y2img, stats);
    gn_apply<<<4096, B256, 0, stream>>>(y2img, stats, spa_gn_g, spa_gn_b, h1, spa);

    // ---- spectral mamba (h1 viewed as [32768,32] sequences of L=4) ----
    gemm_nt_wmma<<<dim3(2048, 2), BW, 0, stream>>>(h1, spe_in_w, nullptr, xz,
                                                   32768, 128, 32, 32, 32, 128);
    conv1d_spectral<<<8192, B256, 0, stream>>>(xz, spe_conv_w, spe_conv_b, xc);
    gemm_nt_wmma<<<dim3(2048, 1), BW, 0, stream>>>(xc, spe_xproj_w, nullptr, proj,
                                                   32768, 34, 64, 64, 64, 48);
    scan_spectral<<<2048, B256, 0, stream>>>(xz, xc, proj, spe_dt_w, spe_dt_b,
                                             spe_Alog, spe_D, yg);
    gemm_nt_wmma<<<dim3(2048, 1), BW, 0, stream>>>(yg, spe_out_w, nullptr, speimg,
                                                   32768, 32, 64, 64, 64, 32);
    gn_stats<<<8, B256, 0, stream>>>(speimg, stats);
    gn_apply<<<4096, B256, 0, stream>>>(speimg, stats, spe_gn_g, spe_gn_b, h1, spe);

    // ---- fuse + heads ----
    fuse_kernel<<<4096, B256, 0, stream>>>(spa, spe, h1, fuse_w, feat);
    gemm_nt_wmma<<<dim3(512, 1), BW, 0, stream>>>(feat, cls_w, cls_b, logits,
                                                  8192, 10, 128, 128, 128, 10);
    dwconv3x3<<<4096, B256, 0, stream>>>(feat, ref_dw_w, edge);
    gn_stats<<<8, B256, 0, stream>>>(edge, stats);
    gn_apply<<<4096, B256, 0, stream>>>(edge, stats, ref_gn_g, ref_gn_b, nullptr, edge);
    pw_sigmoid<<<32, B256, 0, stream>>>(edge, ref_pw_w, ref_pw_b, edge1);
    final_combine<<<320, B256, 0, stream>>>(logits, edge1, alpha, out);
}